// attentionb_37263136260431
// MI455X (gfx1250) — compile-verified
//
#include <hip/hip_runtime.h>
#include <math.h>

// Problem constants (from reference): B=8, CTX=2048, DIMS=1024, HEAD=16, hd=64.
#define NB    8
#define NCTX  2048
#define NDIM  1024
#define NHEAD 16
#define HDD   64
#define NROWS 16            // only seq rows 0..15 are ever non-trivial
#define EPSLN 1e-5f

typedef float v2f __attribute__((ext_vector_type(2)));
typedef float v8f __attribute__((ext_vector_type(8)));

// ---------------------------------------------------------------------------
// f32 WMMA: D(16x16) += A(16x4) x B(4x16), chained over K.
// A layout (per ISA 7.12.2): lanes 0-31 hold M=lane&15; half0 -> K=0,1 ; half1 -> K=2,3.
// B layout (rows striped across lanes): lane holds N=lane&15; half0 -> K=0,1 ; half1 -> K=2,3.
// C/D: lane holds N=lane&15, VGPR r holds M = r + 8*(lane>>4).
// ---------------------------------------------------------------------------
__device__ __forceinline__ v8f wmma_k4(v2f a, v2f b, v8f c) {
  return __builtin_amdgcn_wmma_f32_16x16x4_f32(false, a, false, b, (short)0, c, false, false);
}

// ---------------------------------------------------------------------------
// Zero the 8x2048x1024 output (rows >=16 of every batch stay zero).
// ---------------------------------------------------------------------------
__global__ __launch_bounds__(256) void atb_zero4(float4* __restrict__ p) {
  long i = (long)blockIdx.x * 256 + threadIdx.x;   // grid sized exactly
  p[i] = make_float4(0.f, 0.f, 0.f, 0.f);
}

// ---------------------------------------------------------------------------
// LayerNorm of 16 rows x 1024 per batch -> dst[b][16][1024]
// ---------------------------------------------------------------------------
__global__ __launch_bounds__(128) void atb_ln16(const float* __restrict__ src,
                                                long bstride, long rstride,
                                                const float* __restrict__ g,
                                                const float* __restrict__ bb,
                                                float* __restrict__ dst) {
  __shared__ float red[128];
  const int t = threadIdx.x;
  const int row = blockIdx.x;              // b*16 + r
  const int b = row >> 4, r = row & 15;
  const float* x = src + (long)b * bstride + (long)r * rstride;
  float loc[8];
  float s = 0.f;
#pragma unroll
  for (int j = 0; j < 8; ++j) { loc[j] = x[t + j * 128]; s += loc[j]; }
  red[t] = s; __syncthreads();
  for (int st = 64; st; st >>= 1) { if (t < st) red[t] += red[t + st]; __syncthreads(); }
  const float mean = red[0] * (1.0f / 1024.0f);
  __syncthreads();
  float s2 = 0.f;
#pragma unroll
  for (int j = 0; j < 8; ++j) { float d = loc[j] - mean; s2 += d * d; }
  red[t] = s2; __syncthreads();
  for (int st = 64; st; st >>= 1) { if (t < st) red[t] += red[t + st]; __syncthreads(); }
  const float inv = rsqrtf(red[0] * (1.0f / 1024.0f) + EPSLN);
  float* y = dst + (long)row * NDIM;
#pragma unroll
  for (int j = 0; j < 8; ++j) { int c = t + j * 128; y[c] = (loc[j] - mean) * inv * g[c] + bb[c]; }
}

// ---------------------------------------------------------------------------
// D[b][0:16][cb:cb+16] = A[b](16x1024) @ W(rows cb..cb+15, 1024)^T  via f32 WMMA
// grid = (ntiles, NB), block = 32 (one wave per 16x16 tile, K=1024 chained)
// ---------------------------------------------------------------------------
__global__ __launch_bounds__(32) void atb_gemm16(const float* __restrict__ A,
                                                 const float* __restrict__ W,
                                                 float* __restrict__ D,
                                                 long d_bstride, int d_rstride) {
  const int lane = threadIdx.x;
  const int b = blockIdx.y;
  const int cb = blockIdx.x * 16;
  const int n  = lane & 15;
  const int kh = (lane >> 4) << 1;
  const float* arow = A + (long)b * (NROWS * NDIM) + (long)n * NDIM + kh;  // M = lane&15
  const float* wrow = W + (long)(cb + n) * NDIM + kh;                      // N = lane&15
  __builtin_prefetch(arow, 0, 0);
  __builtin_prefetch(wrow, 0, 0);
  v8f c = {};
#pragma unroll 8
  for (int k0 = 0; k0 < NDIM; k0 += 4) {
    v2f av = *(const v2f*)(arow + k0);
    v2f wv = *(const v2f*)(wrow + k0);
    c = wmma_k4(av, wv, c);
  }
  float* drow = D + (long)b * d_bstride + cb + n;
#pragma unroll
  for (int r = 0; r < 8; ++r)
    drow[(long)(r + ((lane >> 4) << 3)) * d_rstride] = c[r];
}

// ---------------------------------------------------------------------------
// RoPE in place on Q and K buffers [b][16][1024], positions 0..15.
// freq pair p (0..31): fb = 0.2*(21^(p/31)-1); ang = pos * theta/220 * fb
// ---------------------------------------------------------------------------
__global__ __launch_bounds__(256) void atb_rope(float* __restrict__ Qb,
                                                float* __restrict__ Kb,
                                                const float* __restrict__ thetaP) {
  const int gid = blockIdx.x * 256 + threadIdx.x;   // 2*8*16*512 = 131072 exact
  const int p = gid & 511;
  const int rest = gid >> 9;
  const int i = rest & 15;
  const int b = (rest >> 4) & 7;
  const int which = rest >> 7;
  const float theta = thetaP[0];
  const int fi = p & 31;
  const float fb = 0.2f * (powf(21.0f, (float)fi * (1.0f / 31.0f)) - 1.0f);
  const float ang = (float)i * (theta * (1.0f / 220.0f) * fb);
  const float cs = cosf(ang), sn = sinf(ang);
  float* buf = which ? Kb : Qb;
  const long off = (long)b * (NROWS * NDIM) + (long)i * NDIM + 2 * p;
  const float xr = buf[off], xi = buf[off + 1];
  buf[off]     = xr * cs - xi * sn;
  buf[off + 1] = xr * sn + xi * cs;
}

// ---------------------------------------------------------------------------
// dst(16x64) = src(16x64) @ W(64x64)^T + bias, via f32 WMMA (LDS src, global W)
// Must be called with full wave (uniform flow).
// ---------------------------------------------------------------------------
__device__ __forceinline__ void proj64(float* __restrict__ dst, const float* __restrict__ src,
                                       const float* __restrict__ W, const float* __restrict__ bias,
                                       int lane) {
  const int n  = lane & 15;
  const int kh = (lane >> 4) << 1;
  const int mh = (lane >> 4) << 3;
#pragma unroll
  for (int nt = 0; nt < 4; ++nt) {
    v8f c = {};
    const float* arow = src + n * HDD + kh;
    const float* wrow = W + (nt * 16 + n) * HDD + kh;
#pragma unroll
    for (int k0 = 0; k0 < HDD; k0 += 4) {
      v2f av = *(const v2f*)(arow + k0);
      v2f wv = *(const v2f*)(wrow + k0);
      c = wmma_k4(av, wv, c);
    }
    const float bv = bias[nt * 16 + n];
#pragma unroll
    for (int r = 0; r < 8; ++r)
      dst[(r + mh) * HDD + nt * 16 + n] = c[r] + bv;
  }
  __syncthreads();
}

__device__ __forceinline__ void ln_row64(const float* __restrict__ x, float* __restrict__ y,
                                         const float* __restrict__ g, const float* __restrict__ bb) {
  float m = 0.f;
  for (int d = 0; d < HDD; ++d) m += x[d];
  m *= (1.0f / 64.0f);
  float v = 0.f;
  for (int d = 0; d < HDD; ++d) { float t = x[d] - m; v += t * t; }
  v *= (1.0f / 64.0f);
  const float inv = rsqrtf(v + EPSLN);
  for (int d = 0; d < HDD; ++d) y[d] = (x[d] - m) * inv * g[d] + bb[d];
}

// out(16x64) = softmax( LN(qp)*qs @ LN(kp)^T / 8 ) @ vp
__device__ __forceinline__ void sdpa16(const float* __restrict__ qp, const float* __restrict__ kp,
                                       const float* __restrict__ vp, float* __restrict__ out,
                                       float* __restrict__ L1, float* __restrict__ L2,
                                       const float* __restrict__ g, const float* __restrict__ bb,
                                       float qscale, int lane) {
  if (lane < 16) {
    ln_row64(qp + lane * HDD, L1 + lane * HDD, g, bb);
    ln_row64(kp + lane * HDD, L2 + lane * HDD, g, bb);
  }
  __syncthreads();
  if (lane < 16) {
    const int i = lane;
    const float sc = qscale * 0.125f;     // q*s then *d^-0.5 (d=64)
    float lg[16];
    float mx = -3.4e38f;
    for (int j = 0; j < 16; ++j) {
      float d = 0.f;
      for (int t = 0; t < HDD; ++t) d += L1[i * HDD + t] * L2[j * HDD + t];
      d *= sc;
      lg[j] = d; mx = fmaxf(mx, d);
    }
    float se = 0.f;
    for (int j = 0; j < 16; ++j) { lg[j] = expf(lg[j] - mx); se += lg[j]; }
    const float inv = 1.0f / se;
    for (int t = 0; t < HDD; ++t) {
      float acc = 0.f;
      for (int j = 0; j < 16; ++j) acc += lg[j] * vp[j * HDD + t];
      out[i * HDD + t] = acc * inv;
    }
  }
  __syncthreads();
}

// ---------------------------------------------------------------------------
// Iterations 0+1 for one (batch, head). Produces ai1, q+ai0+ai1, kp1, vp1 state
// and the per-unit partial L1-diff sum (fixed-order, deterministic).
// ---------------------------------------------------------------------------
__global__ __launch_bounds__(32) void atb_it01(
    const float* __restrict__ Qb, const float* __restrict__ Kb, const float* __restrict__ Vb,
    const float* __restrict__ Wqh, const float* __restrict__ bqh,
    const float* __restrict__ Wkh, const float* __restrict__ bkh,
    const float* __restrict__ Wvh, const float* __restrict__ bvh,
    const float* __restrict__ lnb_g, const float* __restrict__ lnb_b,
    const float* __restrict__ tempP,
    float* __restrict__ ai1_ws, float* __restrict__ qsum_ws,
    float* __restrict__ kp1_ws, float* __restrict__ vp1_ws,
    float* __restrict__ partial) {
  __shared__ float sA[1024], sB[1024], sC[1024], sD[1024], sE[1024], sF[1024];
  __shared__ float sG[1024], sH[1024], sL1[1024], sL2[1024];
  const int lane = threadIdx.x;
  const int unit = blockIdx.x;             // b*16 + head
  const int b = unit >> 4, hh = unit & 15;
  const long base = (long)b * (NROWS * NDIM) + hh * HDD;
  for (int idx = lane; idx < 1024; idx += 32) {
    const int i = idx >> 6, d = idx & 63;
    const long off = base + (long)i * NDIM + d;
    sA[idx] = Qb[off]; sB[idx] = Kb[off]; sC[idx] = Vb[off];
  }
  __syncthreads();
  const float temp = tempP[0];
  const float qs = (temp != 1.0f && temp > 0.0f) ? sqrtf(1.0f / temp) : 1.0f;
  // it 0
  proj64(sD, sA, Wqh, bqh, lane);                       // qp0
  proj64(sE, sB, Wkh, bkh, lane);                       // kp0
  proj64(sF, sC, Wvh, bvh, lane);                       // vp0
  sdpa16(sD, sE, sF, sG, sL1, sL2, lnb_g, lnb_b, qs, lane);   // ai0
  for (int idx = lane; idx < 1024; idx += 32) sA[idx] += sG[idx];  // curq1 = q + ai0
  __syncthreads();
  // it 1
  proj64(sD, sA, Wqh, bqh, lane);                       // qp1
  proj64(sB, sE, Wkh, bkh, lane);                       // kp1 (from kp0)
  proj64(sC, sF, Wvh, bvh, lane);                       // vp1 (from vp0)
  sdpa16(sD, sB, sC, sH, sL1, sL2, lnb_g, lnb_b, qs, lane);   // ai1
  // deterministic partial |ai1 - ai0|
  float s = 0.f;
  for (int idx = lane; idx < 1024; idx += 32) s += fabsf(sH[idx] - sG[idx]);
#pragma unroll
  for (int m = 16; m; m >>= 1) s += __shfl_xor(s, m, 32);
  if (lane == 0) partial[unit] = s;
  const long ub = (long)unit * 1024;
  for (int idx = lane; idx < 1024; idx += 32) {
    ai1_ws[ub + idx]  = sH[idx];
    qsum_ws[ub + idx] = sA[idx] + sH[idx];              // q + ai0 + ai1
    kp1_ws[ub + idx]  = sB[idx];
    vp1_ws[ub + idx]  = sC[idx];
  }
}

// Global convergence decision: diff1 = sum/denom ; stop iff diff1 < thr + fac*diff1
__global__ __launch_bounds__(128) void atb_decide(const float* __restrict__ partial,
                                                  const float* __restrict__ thrP,
                                                  const float* __restrict__ facP,
                                                  float denom, float* __restrict__ flag) {
  __shared__ float sb[128];
  const int t = threadIdx.x;
  sb[t] = partial[t];
  __syncthreads();
  for (int s = 64; s; s >>= 1) { if (t < s) sb[t] += sb[t + s]; __syncthreads(); }
  if (t == 0) {
    const float diff = sb[0] / denom;
    flag[0] = (diff < thrP[0] + facP[0] * diff) ? 1.0f : 0.0f;
  }
}

// Iteration 2 (or passthrough of ai1 when converged); writes dst[b][16][1024].
__global__ __launch_bounds__(32) void atb_it2(
    const float* __restrict__ ai1_ws, const float* __restrict__ qsum_ws,
    const float* __restrict__ kp1_ws, const float* __restrict__ vp1_ws,
    const float* __restrict__ Wqh, const float* __restrict__ bqh,
    const float* __restrict__ Wkh, const float* __restrict__ bkh,
    const float* __restrict__ Wvh, const float* __restrict__ bvh,
    const float* __restrict__ lnb_g, const float* __restrict__ lnb_b,
    const float* __restrict__ tempP, const float* __restrict__ flag,
    float* __restrict__ dst) {
  __shared__ float sA[1024], sB[1024], sC[1024], sD[1024], sE[1024], sF[1024];
  __shared__ float sG[1024], sL1[1024], sL2[1024];
  const int lane = threadIdx.x;
  const int unit = blockIdx.x;
  const int b = unit >> 4, hh = unit & 15;
  const long ub = (long)unit * 1024;
  const long dbase = (long)b * (NROWS * NDIM) + hh * HDD;
  int stop = (int)flag[0];
  stop = __builtin_amdgcn_readfirstlane(stop);
  if (stop) {                                  // converged: ai2 == ai1
    for (int idx = lane; idx < 1024; idx += 32)
      dst[dbase + (long)(idx >> 6) * NDIM + (idx & 63)] = ai1_ws[ub + idx];
    return;
  }
  for (int idx = lane; idx < 1024; idx += 32) {
    sA[idx] = qsum_ws[ub + idx];
    sB[idx] = kp1_ws[ub + idx];
    sC[idx] = vp1_ws[ub + idx];
  }
  __syncthreads();
  const float temp = tempP[0];
  const float qs = (temp != 1.0f && temp > 0.0f) ? sqrtf(1.0f / temp) : 1.0f;
  proj64(sD, sA, Wqh, bqh, lane);
  proj64(sE, sB, Wkh, bkh, lane);
  proj64(sF, sC, Wvh, bvh, lane);
  sdpa16(sD, sE, sF, sG, sL1, sL2, lnb_g, lnb_b, qs, lane);
  for (int idx = lane; idx < 1024; idx += 32)
    dst[dbase + (long)(idx >> 6) * NDIM + (idx & 63)] = sG[idx];
}

// ---------------------------------------------------------------------------
extern "C" void kernel_launch(void* const* d_in, const int* in_sizes, int n_in,
                              void* d_out, int out_size, void* d_ws, size_t ws_size,
                              hipStream_t stream) {
  const float* x     = (const float*)d_in[0];
  const float* xa    = (const float*)d_in[1];
  const float* Wq    = (const float*)d_in[2];
  const float* Wkv   = (const float*)d_in[3];
  const float* Wout  = (const float*)d_in[4];
  const float* lna_g = (const float*)d_in[5];
  const float* lna_b = (const float*)d_in[6];
  const float* lnb_g = (const float*)d_in[7];
  const float* lnb_b = (const float*)d_in[8];
  const float* Wqh   = (const float*)d_in[9];
  const float* bqh   = (const float*)d_in[10];
  const float* Wkh   = (const float*)d_in[11];
  const float* bkh   = (const float*)d_in[12];
  const float* Wvh   = (const float*)d_in[13];
  const float* bvh   = (const float*)d_in[14];
  const float* theta = (const float*)d_in[15];
  const float* thr   = (const float*)d_in[16];
  const float* fac   = (const float*)d_in[17];
  const float* temp  = (const float*)d_in[18];
  float* out = (float*)d_out;

  const long U = NB * NROWS * NDIM;            // 131072 floats per [8][16][1024] buffer
  float* w    = (float*)d_ws;
  float* Hq   = w;          float* Hkv  = w + 1 * U;
  float* Qb   = w + 2 * U;  float* Kb   = w + 3 * U;  float* Vb = w + 4 * U;
  float* Y16  = w + 5 * U;  float* YA16 = w + 6 * U;  float* F3O = w + 7 * U;
  float* AI1  = w + 8 * U;  float* QSUM = w + 9 * U;
  float* KP1  = w + 10 * U; float* VP1  = w + 11 * U;
  float* PART = w + 12 * U; float* FLAG = PART + 128;

  // Zero full output (only rows 0..15 of each batch get overwritten below).
  atb_zero4<<<(NB * NCTX * NDIM) / (4 * 256), 256, 0, stream>>>((float4*)out);

  auto focus = [&](const float* qsrc, long qb, long qr,
                   const float* kvsrc, long kvb, long kvr,
                   float denom, float* dst) {
    atb_ln16<<<NB * NROWS, 128, 0, stream>>>(qsrc, qb, qr, lna_g, lna_b, Hq);
    const float* Hk = Hq;
    if (kvsrc != qsrc) {
      atb_ln16<<<NB * NROWS, 128, 0, stream>>>(kvsrc, kvb, kvr, lna_g, lna_b, Hkv);
      Hk = Hkv;
    }
    atb_gemm16<<<dim3(64, NB), 32, 0, stream>>>(Hq, Wq,                     Qb, NROWS * NDIM, NDIM);
    atb_gemm16<<<dim3(64, NB), 32, 0, stream>>>(Hk, Wkv,                    Kb, NROWS * NDIM, NDIM);
    atb_gemm16<<<dim3(64, NB), 32, 0, stream>>>(Hk, Wkv + (long)NDIM * NDIM, Vb, NROWS * NDIM, NDIM);
    atb_rope<<<512, 256, 0, stream>>>(Qb, Kb, theta);
    atb_it01<<<NB * NHEAD, 32, 0, stream>>>(Qb, Kb, Vb, Wqh, bqh, Wkh, bkh, Wvh, bvh,
                                            lnb_g, lnb_b, temp, AI1, QSUM, KP1, VP1, PART);
    atb_decide<<<1, 128, 0, stream>>>(PART, thr, fac, denom, FLAG);
    atb_it2<<<NB * NHEAD, 32, 0, stream>>>(AI1, QSUM, KP1, VP1, Wqh, bqh, Wkh, bkh, Wvh, bvh,
                                           lnb_g, lnb_b, temp, FLAG, dst);
  };

  // Stage 1: y16  = focus(x[:, :64], x[:, :64]) rows 0..15   (diff denom cq=64)
  focus(x,  (long)NCTX * NDIM, NDIM, x,  (long)NCTX * NDIM, NDIM,
        (float)(NB * NHEAD * 64 * HDD), Y16);
  // Stage 2: ya16 = focus(xa[:, :64], xa[:, :64]) rows 0..15
  focus(xa, (long)NCTX * NDIM, NDIM, xa, (long)NCTX * NDIM, NDIM,
        (float)(NB * NHEAD * 64 * HDD), YA16);
  // Stage 3: final focus(y, ya) rows 0..15   (diff denom cq=2048)
  focus(Y16, (long)NROWS * NDIM, NDIM, YA16, (long)NROWS * NDIM, NDIM,
        (float)(NB * NHEAD * 2048 * HDD), F3O);
  // Stage 4: out[b, 0:16, :] = F3O @ Wout^T
  atb_gemm16<<<dim3(64, NB), 32, 0, stream>>>(F3O, Wout, out, (long)NCTX * NDIM, NDIM);
}